// Alignment_loss_34196529611087
// MI455X (gfx1250) — compile-verified
//
#include <hip/hip_runtime.h>
#include <hip/hip_bf16.h>
#include <math.h>

typedef float v2f __attribute__((ext_vector_type(2)));
typedef float v8f __attribute__((ext_vector_type(8)));

#define NROWS 1024
#define DDIM  128
// scale = 1/(TAU*sqrt(D)) = 1/sqrt(128)
#define SCALE 0.08838834764831845f

#define CHUNKS 8
#define TILES_PER_CHUNK 8   // 64 column tiles total / 8 chunks

// workspace layout (float offsets)
#define OFF_NX    0
#define OFF_NP    1024
#define OFF_NA    2048
#define OFF_POSC  3072
#define OFF_POSI  4096
#define OFF_PARTC 5120
#define OFF_PARTI (5120 + CHUNKS * NROWS)
// total floats = 5120 + 2*8*1024 = 21504  (84 KB)

// ---------------------------------------------------------------------------
// Kernel 1: per-row squared norms of x, p(label), a(aug) and the diagonal
// "pos" distances.  One wave per row, each lane loads a float4 (128/32 = 4).
// ---------------------------------------------------------------------------
__global__ __launch_bounds__(256) void rownorm_kernel(
    const float* __restrict__ x, const float* __restrict__ p,
    const float* __restrict__ a, float* __restrict__ ws) {
  const int wave = threadIdx.x >> 5;
  const int lane = threadIdx.x & 31;
  const int row  = blockIdx.x * 8 + wave;

  const float4 xv = ((const float4*)(x + row * DDIM))[lane];
  const float4 pv = ((const float4*)(p + row * DDIM))[lane];
  const float4 av = ((const float4*)(a + row * DDIM))[lane];

  float sx = xv.x * xv.x + xv.y * xv.y + xv.z * xv.z + xv.w * xv.w;
  float sp = pv.x * pv.x + pv.y * pv.y + pv.z * pv.z + pv.w * pv.w;
  float sa = av.x * av.x + av.y * av.y + av.z * av.z + av.w * av.w;
  float dxp = (xv.x - pv.x) * (xv.x - pv.x) + (xv.y - pv.y) * (xv.y - pv.y) +
              (xv.z - pv.z) * (xv.z - pv.z) + (xv.w - pv.w) * (xv.w - pv.w);
  float dxa = (xv.x - av.x) * (xv.x - av.x) + (xv.y - av.y) * (xv.y - av.y) +
              (xv.z - av.z) * (xv.z - av.z) + (xv.w - av.w) * (xv.w - av.w);

  #pragma unroll
  for (int m = 16; m >= 1; m >>= 1) {
    sx  += __shfl_xor(sx, m, 32);
    sp  += __shfl_xor(sp, m, 32);
    sa  += __shfl_xor(sa, m, 32);
    dxp += __shfl_xor(dxp, m, 32);
    dxa += __shfl_xor(dxa, m, 32);
  }
  if (lane == 0) {
    ws[OFF_NX + row]   = sx;
    ws[OFF_NP + row]   = sp;
    ws[OFF_NA + row]   = sa;
    ws[OFF_POSC + row] = sqrtf(dxp) * SCALE;
    ws[OFF_POSI + row] = sqrtf(dxa) * SCALE;
  }
}

// ---------------------------------------------------------------------------
// Kernel 2: Gram tiles via V_WMMA_F32_16X16X4_F32 (exact fp32), converted to
// exp(distance) row partial sums.  One wave (32 threads) per block.
// blockIdx.x = rowtile * CHUNKS + chunk.
//
// fp32 WMMA fragment layout (wave32):
//   A 16x4 : lane 0-15 -> row M=lane,  VGPR0=K0, VGPR1=K1;
//            lane 16-31 -> same rows,  VGPR0=K2, VGPR1=K3.
//   B 4x16 : same with N=lane (i.e. rows of Y for X*Y^T).
//   C 16x16: VGPR v: lanes 0-15 -> (M=v,   N=lane),
//                    lanes 16-31-> (M=v+8, N=lane-16).
// ---------------------------------------------------------------------------
__global__ __launch_bounds__(32) void pair_tile_kernel(
    const float* __restrict__ X, const float* __restrict__ Y,
    const float* __restrict__ nx, const float* __restrict__ ny,
    float* __restrict__ part /* [CHUNKS][NROWS] */) {
  const int rowtile = blockIdx.x / CHUNKS;
  const int chunk   = blockIdx.x % CHUNKS;
  const int lane    = threadIdx.x;
  const int half    = lane >> 4;    // 0 or 1
  const int l16     = lane & 15;
  const int rowbase = rowtile * 16;
  const int koff    = 2 * half;     // K offset for this lane half

  // A fragments: persistent across all 8 column tiles of this chunk.
  v2f afrag[32];
  const float* arow = X + (rowbase + l16) * DDIM + koff;
  #pragma unroll
  for (int kk = 0; kk < 32; ++kk)
    afrag[kk] = *(const v2f*)(arow + kk * 4);

  // ||x||^2 for the 8 C-rows this lane touches: rowbase + half*8 + v
  float nxr[8];
  #pragma unroll
  for (int v = 0; v < 8; ++v) nxr[v] = nx[rowbase + half * 8 + v];

  float rsum[8];
  #pragma unroll
  for (int v = 0; v < 8; ++v) rsum[v] = 0.0f;

  for (int t = 0; t < TILES_PER_CHUNK; ++t) {
    const int colbase = (chunk * TILES_PER_CHUNK + t) * 16;
    const float* brow = Y + (colbase + l16) * DDIM + koff;
    v2f bfrag[32];
    #pragma unroll
    for (int kk = 0; kk < 32; ++kk)
      bfrag[kk] = *(const v2f*)(brow + kk * 4);

    v8f acc = {};
    #pragma unroll
    for (int kk = 0; kk < 32; ++kk)
      acc = __builtin_amdgcn_wmma_f32_16x16x4_f32(
          /*neg_a=*/false, afrag[kk], /*neg_b=*/false, bfrag[kk],
          /*c_mod=*/(short)0, acc, /*reuse_a=*/false, /*reuse_b=*/false);

    const float nyv = ny[colbase + l16];
    #pragma unroll
    for (int v = 0; v < 8; ++v) {
      float d2 = nxr[v] + nyv - 2.0f * acc[v];
      d2 = fmaxf(d2, 0.0f);
      rsum[v] += expf(sqrtf(d2) * SCALE);
    }
  }

  // Reduce the 16 column-lanes of each half; lane 0 / lane 16 own the rows.
  #pragma unroll
  for (int v = 0; v < 8; ++v) {
    float s = rsum[v];
    s += __shfl_xor(s, 1, 16);
    s += __shfl_xor(s, 2, 16);
    s += __shfl_xor(s, 4, 16);
    s += __shfl_xor(s, 8, 16);
    if (l16 == 0)
      part[chunk * NROWS + rowbase + half * 8 + v] = s;
  }
}

// ---------------------------------------------------------------------------
// Kernel 3: denom = sum over chunks; loss = mean(pos - log(denom)); 3 outputs.
// ---------------------------------------------------------------------------
__global__ __launch_bounds__(256) void finalize_kernel(
    const float* __restrict__ ws, float* __restrict__ out) {
  __shared__ float sc[256];
  __shared__ float si[256];
  const int tid = threadIdx.x;
  float lc = 0.0f, li = 0.0f;
  for (int i = tid; i < NROWS; i += 256) {
    float dc = 0.0f, di = 0.0f;
    #pragma unroll
    for (int c = 0; c < CHUNKS; ++c) {
      dc += ws[OFF_PARTC + c * NROWS + i];
      di += ws[OFF_PARTI + c * NROWS + i];
    }
    lc += ws[OFF_POSC + i] - logf(dc);
    li += ws[OFF_POSI + i] - logf(di);
  }
  sc[tid] = lc;
  si[tid] = li;
  __syncthreads();
  for (int s = 128; s > 0; s >>= 1) {
    if (tid < s) {
      sc[tid] += sc[tid + s];
      si[tid] += si[tid + s];
    }
    __syncthreads();
  }
  if (tid == 0) {
    const float center   = sc[0] * (1.0f / NROWS);
    const float instance = si[0] * (1.0f / NROWS);
    out[0] = center + instance;  // alignment_loss (BETA = 1)
    out[1] = center;             // center_alignment_loss
    out[2] = instance;           // instance_alignment_loss
  }
}

extern "C" void kernel_launch(void* const* d_in, const int* in_sizes, int n_in,
                              void* d_out, int out_size, void* d_ws, size_t ws_size,
                              hipStream_t stream) {
  const float* x   = (const float*)d_in[0];  // "x"
  const float* aug = (const float*)d_in[1];  // "aug_x"
  const float* lab = (const float*)d_in[2];  // "label_prompt_embedding"
  float* ws  = (float*)d_ws;
  float* out = (float*)d_out;

  // 1) row norms + diagonal pos distances
  rownorm_kernel<<<NROWS / 8, 256, 0, stream>>>(x, lab, aug, ws);

  // 2) pairwise exp-distance row partial sums via fp32 WMMA
  pair_tile_kernel<<<(NROWS / 16) * CHUNKS, 32, 0, stream>>>(
      x, lab, ws + OFF_NX, ws + OFF_NP, ws + OFF_PARTC);
  pair_tile_kernel<<<(NROWS / 16) * CHUNKS, 32, 0, stream>>>(
      x, aug, ws + OFF_NX, ws + OFF_NA, ws + OFF_PARTI);

  // 3) final reduction to the three scalars
  finalize_kernel<<<1, 256, 0, stream>>>(ws, out);
}